// RegressionLoss_33423435497886
// MI455X (gfx1250) — compile-verified
//
#include <hip/hip_runtime.h>
#include <hip/hip_bf16.h>

// ---------------------------------------------------------------------------
// RegressionLoss on MI455X (gfx1250, wave32).
// num/den accumulated per (b, pixel-chunk) block; per-wave lane reduction done
// with V_WMMA_F32_16X16X4_F32 (B = ones) in full f32 precision; deterministic
// two-stage reduction (no float atomics) so graph replays are bit-identical.
// ---------------------------------------------------------------------------

#define BB 16
#define TT 16
#define HH 320
#define WW 320
#define HW (HH * WW)            // 102400
#define THREADS 256
#define PPT 4                   // pixels per thread
#define PPB (THREADS * PPT)     // 1024 pixels per block
#define BLOCKS_PER_B (HW / PPB) // 100
#define EPSF 1e-8f
#define THETA_COEF 10.0f

typedef float v2f __attribute__((ext_vector_type(2)));
typedef float v8f __attribute__((ext_vector_type(8)));

// Exact f32 sum of one value across all 32 lanes of a wave, via WMMA.
// A(16x4): lane L<16 holds A[L][0]=x, A[L][1]=0; lane L+16 holds A[L][2]=x, A[L][3]=0.
// B = all ones (layout-independent).  D[m][*] = x[m] + x[m+16].
// D VGPR k: lanes0-15 = D[k][n], lanes16-31 = D[k+8][n].  Sum 8 VGPRs, then
// xor-16 shuffle-add -> full 32-lane sum in every lane.
__device__ __forceinline__ float wave_sum_wmma(float x) {
    v2f a; a[0] = x;   a[1] = 0.0f;
    v2f o; o[0] = 1.0f; o[1] = 1.0f;
    v8f c = {0.f, 0.f, 0.f, 0.f, 0.f, 0.f, 0.f, 0.f};
    v8f d = __builtin_amdgcn_wmma_f32_16x16x4_f32(
        /*neg_a=*/false, a, /*neg_b=*/false, o,
        /*c_mod=*/(short)0, c, /*reuse_a=*/false, /*reuse_b=*/false);
    float s = d[0] + d[1] + d[2] + d[3] + d[4] + d[5] + d[6] + d[7];
    s += __shfl_xor(s, 16, 32);
    return s;
}

__global__ void __launch_bounds__(THREADS)
loss_main_kernel(const float* __restrict__ boxes,   // (B,H,W,4)
                 const float* __restrict__ rects,   // (B,T,12)
                 const float* __restrict__ pth,     // (B,H,W,1)
                 const float* __restrict__ tth,     // (B,T,1)
                 float* __restrict__ ws)            // (B, BLOCKS_PER_B, 2)
{
    __shared__ float sc[TT][18];  // per-target constants
    __shared__ float swn[THREADS / 32];
    __shared__ float swd[THREADS / 32];

    const int b   = blockIdx.y;
    const int tid = threadIdx.x;

    // ---- per-target constants (affine edge coefficients, box, area, theta) ----
    if (tid < TT) {
        const float* r = rects + (b * TT + tid) * 12;
        float txmin = r[0], tymin = r[1], txmax = r[2], tymax = r[3];
        float qx[4] = {r[4], r[6], r[8], r[10]};
        float qy[4] = {r[5], r[7], r[9], r[11]};
        float* s = sc[tid];
        s[0] = txmin; s[1] = tymin; s[2] = txmax; s[3] = tymax;
        s[4] = (txmax - txmin) * (tymax - tymin) + EPSF;   // area_t + eps folded
        s[5] = tth[b * TT + tid];
        #pragma unroll
        for (int e = 0; e < 4; ++e) {
            float ex = qx[(e + 1) & 3] - qx[e];
            float ey = qy[(e + 1) & 3] - qy[e];
            // cross = ex*py - ey*px + (ey*qx - ex*qy)
            s[6 + 3 * e] = ex;
            s[7 + 3 * e] = -ey;
            s[8 + 3 * e] = ey * qx[e] - ex * qy[e];
        }
    }
    __syncthreads();

    // ---- load this thread's 4 pixels (float4 box, theta, coords, area_p) ----
    const int base = blockIdx.x * PPB;
    float pxmin[PPT], pymin[PPT], pxmax[PPT], pymax[PPT];
    float areap[PPT], ptv[PPT], fx[PPT], fy[PPT];
    #pragma unroll
    for (int i = 0; i < PPT; ++i) {
        int p = base + i * THREADS + tid;
        const float4 pb = ((const float4*)boxes)[(size_t)b * HW + p];
        pxmin[i] = pb.x; pymin[i] = pb.y; pxmax[i] = pb.z; pymax[i] = pb.w;
        areap[i] = fmaxf(pb.z - pb.x, 0.f) * fmaxf(pb.w - pb.y, 0.f);
        ptv[i]   = pth[(size_t)b * HW + p];
        int h = p / WW;
        fx[i] = (float)(p - h * WW);
        fy[i] = (float)h;
    }

    float num = 0.f, den = 0.f;

    // ---- t outer (constants in registers), pixels inner ----
    for (int t = 0; t < TT; ++t) {
        const float* s = sc[t];
        const float txmin = s[0], tymin = s[1], txmax = s[2], tymax = s[3];
        const float den_t = s[4];       // area_t + eps
        const float thv   = s[5];
        const float e0a = s[6],  e0b = s[7],  e0c = s[8];
        const float e1a = s[9],  e1b = s[10], e1c = s[11];
        const float e2a = s[12], e2b = s[13], e2c = s[14];
        const float e3a = s[15], e3b = s[16], e3c = s[17];
        #pragma unroll
        for (int i = 0; i < PPT; ++i) {
            // convex-quad containment: all crosses >= 0 or all <= 0
            float c0 = fmaf(e0a, fy[i], fmaf(e0b, fx[i], e0c));
            float c1 = fmaf(e1a, fy[i], fmaf(e1b, fx[i], e1c));
            float c2 = fmaf(e2a, fy[i], fmaf(e2b, fx[i], e2c));
            float c3 = fmaf(e3a, fy[i], fmaf(e3b, fx[i], e3c));
            float mn = fminf(fminf(c0, c1), fminf(c2, c3));
            float mx = fmaxf(fmaxf(c0, c1), fmaxf(c2, c3));
            float m  = (mn >= 0.f || mx <= 0.f) ? 1.f : 0.f;
            // dice
            float iw = fmaxf(fminf(pxmax[i], txmax) - fmaxf(pxmin[i], txmin), 0.f);
            float ih = fmaxf(fminf(pymax[i], tymax) - fmaxf(pymin[i], tymin), 0.f);
            float inter = iw * ih;
            float rden  = __builtin_amdgcn_rcpf(areap[i] + den_t);
            float dice  = fmaf(-2.f * inter, rden, 1.f);
            // orientation cosine: c / max(|c|, eps)
            float cth  = ptv[i] * thv;
            float cosv = cth * __builtin_amdgcn_rcpf(fmaxf(fabsf(cth), EPSF));
            num = fmaf(m, fmaf(THETA_COEF, cosv, dice), num);
            den += m;
        }
    }

    // ---- wave reduction via WMMA (all lanes converged -> EXEC all ones) ----
    float wn = wave_sum_wmma(num);
    float wd = wave_sum_wmma(den);

    const int wave = tid >> 5;
    if ((tid & 31) == 0) { swn[wave] = wn; swd[wave] = wd; }
    __syncthreads();

    if (tid == 0) {
        float bn = 0.f, bd = 0.f;
        #pragma unroll
        for (int w = 0; w < THREADS / 32; ++w) { bn += swn[w]; bd += swd[w]; }
        size_t slot = ((size_t)b * BLOCKS_PER_B + blockIdx.x) * 2;
        ws[slot + 0] = bn;   // overwritten every launch -> deterministic
        ws[slot + 1] = bd;
    }
}

__global__ void loss_finalize_kernel(const float* __restrict__ ws,
                                     float* __restrict__ out)
{
    int b = threadIdx.x;
    if (b < BB) {
        float n = 0.f, d = 0.f;
        for (int i = 0; i < BLOCKS_PER_B; ++i) {
            n += ws[((size_t)b * BLOCKS_PER_B + i) * 2 + 0];
            d += ws[((size_t)b * BLOCKS_PER_B + i) * 2 + 1];
        }
        out[b] = n / d;
    }
}

extern "C" void kernel_launch(void* const* d_in, const int* in_sizes, int n_in,
                              void* d_out, int out_size, void* d_ws, size_t ws_size,
                              hipStream_t stream) {
    // setup_inputs order: pred_confs, pred_boxes, true_rects, pred_thetas, true_thetas
    const float* pred_boxes  = (const float*)d_in[1];
    const float* true_rects  = (const float*)d_in[2];
    const float* pred_thetas = (const float*)d_in[3];
    const float* true_thetas = (const float*)d_in[4];
    float* out = (float*)d_out;
    float* ws  = (float*)d_ws;   // needs B*BLOCKS_PER_B*2*4 = 12.8 KB

    dim3 grid(BLOCKS_PER_B, BB);
    loss_main_kernel<<<grid, THREADS, 0, stream>>>(pred_boxes, true_rects,
                                                   pred_thetas, true_thetas, ws);
    loss_finalize_kernel<<<1, 32, 0, stream>>>(ws, out);
    (void)in_sizes; (void)n_in; (void)out_size; (void)ws_size;
}